// ArFSSM_77103252898296
// MI455X (gfx1250) — compile-verified
//
#include <hip/hip_runtime.h>
#include <math.h>

// Problem constants
#define BB   256
#define TT   512
#define DXC  100
#define DZC  100
#define DH1C 10
#define DH2C 10
#define DDZC 20
#define DDXC 20
#define G1W  30   // 3*DH1
#define G2W  30   // 3*DH2
#define GPAD 32   // padded gate width
#define OUTW (2 * DXC + 3 * DZC)   // 500

typedef __attribute__((ext_vector_type(2))) float v2f;
typedef __attribute__((ext_vector_type(8))) float v8f;

__device__ __forceinline__ v8f wmma_f32_k4(v2f a, v2f b, v8f c) {
    // D = A(16x4) * B(4x16) + C, fp32 WMMA
    return __builtin_amdgcn_wmma_f32_16x16x4_f32(
        /*neg_a=*/false, a, /*neg_b=*/false, b,
        /*c_mod=*/(short)0, c, /*reuse_a=*/false, /*reuse_b=*/false);
}

__device__ __forceinline__ float sigmoidf_(float x) { return 1.0f / (1.0f + __expf(-x)); }
__device__ __forceinline__ float softplusf_(float x) { return (x > 20.0f) ? x : log1pf(__expf(x)); }

// -----------------------------------------------------------------------------
// Single 16x16 N-tile of C = A(16xKpad) @ W(N x Kreal)^T + bias, one wave.
// A row-major (LDS, zero-padded Kreal..Kpad), W/bias LDS or global, C to LDS.
// -----------------------------------------------------------------------------
__device__ __forceinline__ void gemm_tile(const float* W, const float* bias,
                                          const float* A, int lda,
                                          float* C, int ldc,
                                          int Kreal, int Kpad, int nb, int N, int lane)
{
    const int half = lane >> 4;
    const int l15  = lane & 15;
    const int koff = half * 2;
    const int mbase = half * 8;
    const int col = nb + l15;
    const bool cv = (col < N);
    const float bv = cv ? bias[col] : 0.0f;
    v8f c;
#pragma unroll
    for (int i = 0; i < 8; ++i) c[i] = bv;
    for (int k = 0; k < Kpad; k += 4) {
        v2f a, b;
        a[0] = A[l15 * lda + k + koff];
        a[1] = A[l15 * lda + k + koff + 1];
        float b0 = 0.0f, b1 = 0.0f;
        if (cv) {
            if (k + koff     < Kreal) b0 = W[col * Kreal + k + koff];
            if (k + koff + 1 < Kreal) b1 = W[col * Kreal + k + koff + 1];
        }
        b[0] = b0; b[1] = b1;
        c = wmma_f32_k4(a, b, c);
    }
    if (cv) {
#pragma unroll
        for (int i = 0; i < 8; ++i) C[(mbase + i) * ldc + col] = c[i];
    }
}

// -----------------------------------------------------------------------------
// Kernel A: gi2 = x @ W_ih2^T + b_ih2 for all B*T rows.  (131072 x 100) x (100 x 30)
// One wave per 16-row M tile; stores padded stride-32 rows into workspace.
// -----------------------------------------------------------------------------
__global__ void __launch_bounds__(256) k_gi2(const float* __restrict__ x,
                                             const float* __restrict__ W,
                                             const float* __restrict__ bias,
                                             float* __restrict__ gi2)
{
    const int gwave = (int)((blockIdx.x * blockDim.x + threadIdx.x) >> 5);
    const int lane  = (int)(threadIdx.x & 31);
    const long row0 = (long)gwave * 16;
    const int half = lane >> 4;
    const int l15  = lane & 15;
    const int koff = half * 2;
    const int mbase = half * 8;
    const float* xrow = x + (row0 + l15) * DXC;

    for (int nb = 0; nb < G2W; nb += 16) {
        const int col = nb + l15;
        const bool cv = (col < G2W);
        const float bv = cv ? bias[col] : 0.0f;
        v8f c;
#pragma unroll
        for (int i = 0; i < 8; ++i) c[i] = bv;
        for (int k = 0; k < DXC; k += 4) {
            v2f a, b;
            a[0] = xrow[k + koff];
            a[1] = xrow[k + koff + 1];
            b[0] = cv ? W[col * DXC + k + koff]     : 0.0f;
            b[1] = cv ? W[col * DXC + k + koff + 1] : 0.0f;
            c = wmma_f32_k4(a, b, c);
        }
        if (cv) {
#pragma unroll
            for (int i = 0; i < 8; ++i) gi2[(row0 + mbase + i) * GPAD + col] = c[i];
        }
    }
}

// -----------------------------------------------------------------------------
// Kernel B: GRU2 scan. One lane per batch element, h2 (10) in registers,
// sequential over T. Stores the SHIFTED h2 (state before consuming x_t).
// -----------------------------------------------------------------------------
__global__ void __launch_bounds__(128) k_gru2(const float* __restrict__ gi2,
                                              const float* __restrict__ Whh,
                                              const float* __restrict__ bhh,
                                              const float* __restrict__ h2_0,
                                              float* __restrict__ h2s)
{
    const int b = (int)(blockIdx.x * blockDim.x + threadIdx.x);
    if (b >= BB) return;
    float h[DH2C];
#pragma unroll
    for (int j = 0; j < DH2C; ++j) h[j] = h2_0[j];

    for (int t = 0; t < TT; ++t) {
        const long r = (long)b * TT + t;
        float* hs = h2s + r * DH2C;
#pragma unroll
        for (int j = 0; j < DH2C; ++j) hs[j] = h[j];

        const float* gi = gi2 + r * GPAD;
        float gh[G2W];
#pragma unroll
        for (int g = 0; g < G2W; ++g) {
            float s = bhh[g];
#pragma unroll
            for (int j = 0; j < DH2C; ++j) s += Whh[g * DH2C + j] * h[j];
            gh[g] = s;
        }
#pragma unroll
        for (int j = 0; j < DH2C; ++j) {
            const float rg = sigmoidf_(gi[j] + gh[j]);
            const float ug = sigmoidf_(gi[DH2C + j] + gh[DH2C + j]);
            const float ng = tanhf(gi[2 * DH2C + j] + rg * gh[2 * DH2C + j]);
            h[j] = (1.0f - ug) * ng + ug * h[j];
        }
    }
}

// -----------------------------------------------------------------------------
// One 16-wide N-tile of the dz heads: loc/scale GEMMs + softplus + reparam
// sample; z -> LDS, (z_loc, z_scale, z_t) -> global out.
// -----------------------------------------------------------------------------
__device__ __forceinline__ void dz_head_tile(const float* Wloc, const float* bloc,
                                             const float* __restrict__ Wscale,
                                             const float* bscale,
                                             const float* G,      // LDS 16x20
                                             const float* epsl,   // LDS 16x100
                                             float* zlds,         // LDS 16x100
                                             float* __restrict__ out,
                                             int b0, int t, int nb, int lane)
{
    const int half = lane >> 4, l15 = lane & 15;
    const int koff = half * 2, mbase = half * 8;
    const int col = nb + l15;
    const bool cv = (col < DZC);
    const float bl = cv ? bloc[col] : 0.0f;
    const float bs = cv ? bscale[col] : 0.0f;
    v8f cl, cs;
#pragma unroll
    for (int i = 0; i < 8; ++i) { cl[i] = bl; cs[i] = bs; }
    for (int k = 0; k < DDZC; k += 4) {
        v2f a, wl, ws;
        a[0] = G[l15 * DDZC + k + koff];
        a[1] = G[l15 * DDZC + k + koff + 1];
        wl[0] = cv ? Wloc[col * DDZC + k + koff]       : 0.0f;
        wl[1] = cv ? Wloc[col * DDZC + k + koff + 1]   : 0.0f;
        ws[0] = cv ? Wscale[col * DDZC + k + koff]     : 0.0f;
        ws[1] = cv ? Wscale[col * DDZC + k + koff + 1] : 0.0f;
        cl = wmma_f32_k4(a, wl, cl);
        cs = wmma_f32_k4(a, ws, cs);
    }
    if (cv) {
#pragma unroll
        for (int i = 0; i < 8; ++i) {
            const int m = mbase + i;
            const size_t r = (size_t)(b0 + m) * TT + t;
            const float loc = cl[i];
            const float sc  = softplusf_(cs[i]);
            const float z   = loc + sc * epsl[m * DZC + col];
            zlds[m * DZC + col] = z;
            float* o = out + r * OUTW;
            o[2 * DXC + col]           = loc;  // z_loc
            o[2 * DXC + DZC + col]     = sc;   // z_scale
            o[2 * DXC + 2 * DZC + col] = z;    // z_t
        }
    }
}

// One 16-wide N-tile of the dx heads: (x_loc, x_scale) -> global out.
__device__ __forceinline__ void dx_head_tile(const float* Wloc, const float* bloc,
                                             const float* __restrict__ Wscale,
                                             const float* bscale,
                                             const float* G,      // LDS 16x20
                                             float* __restrict__ out,
                                             int b0, int t, int nb, int lane)
{
    const int half = lane >> 4, l15 = lane & 15;
    const int koff = half * 2, mbase = half * 8;
    const int col = nb + l15;
    const bool cv = (col < DXC);
    const float bl = cv ? bloc[col] : 0.0f;
    const float bs = cv ? bscale[col] : 0.0f;
    v8f cl, cs;
#pragma unroll
    for (int i = 0; i < 8; ++i) { cl[i] = bl; cs[i] = bs; }
    for (int k = 0; k < DDXC; k += 4) {
        v2f a, wl, ws;
        a[0] = G[l15 * DDXC + k + koff];
        a[1] = G[l15 * DDXC + k + koff + 1];
        wl[0] = cv ? Wloc[col * DDXC + k + koff]       : 0.0f;
        wl[1] = cv ? Wloc[col * DDXC + k + koff + 1]   : 0.0f;
        ws[0] = cv ? Wscale[col * DDXC + k + koff]     : 0.0f;
        ws[1] = cv ? Wscale[col * DDXC + k + koff + 1] : 0.0f;
        cl = wmma_f32_k4(a, wl, cl);
        cs = wmma_f32_k4(a, ws, cs);
    }
    if (cv) {
#pragma unroll
        for (int i = 0; i < 8; ++i) {
            const int m = mbase + i;
            const size_t r = (size_t)(b0 + m) * TT + t;
            float* o = out + r * OUTW;
            o[col]       = cl[i];               // x_loc
            o[DXC + col] = softplusf_(cs[i]);   // x_scale
        }
    }
}

// LDS weight-cache offsets (floats)
#define OFF_DZW1   0        // 200
#define OFF_DZB1   200      // 20
#define OFF_DZWLOC 220      // 2000
#define OFF_DZBLOC 2220     // 100
#define OFF_DZBSC  2320     // 100
#define OFF_WIH1   2420     // 3000
#define OFF_BIH1   5420     // 30
#define OFF_WHH1   5450     // 300
#define OFF_BHH1   5750     // 30
#define OFF_DXW1   5780     // 400
#define OFF_DXB1   6180     // 20
#define OFF_DXWLOC 6200     // 2000
#define OFF_DXBLOC 8200     // 100
#define OFF_DXBSC  8300     // 100
#define WT_TOTAL   8400     // 33.6 KB

__device__ __forceinline__ void ldcpy(float* dst, const float* __restrict__ src,
                                      int n, int tid, int nt) {
    for (int i = tid; i < n; i += nt) dst[i] = src[i];
}

// -----------------------------------------------------------------------------
// Kernel C: sequential generative rollout. 4 waves per 16-row batch tile.
// Recurrence-critical weights cached in LDS; GEMM stages split across waves;
// eps / h2_shift prefetched into LDS off the critical path.
// -----------------------------------------------------------------------------
__global__ void __launch_bounds__(128) k_main(
    const float* __restrict__ eps,
    const float* __restrict__ W_ih1, const float* __restrict__ W_hh1,
    const float* __restrict__ b_ih1, const float* __restrict__ b_hh1,
    const float* __restrict__ dz_W1, const float* __restrict__ dz_b1,
    const float* __restrict__ dz_Wloc, const float* __restrict__ dz_bloc,
    const float* __restrict__ dz_Wscale, const float* __restrict__ dz_bscale,
    const float* __restrict__ dx_W1, const float* __restrict__ dx_b1,
    const float* __restrict__ dx_Wloc, const float* __restrict__ dx_bloc,
    const float* __restrict__ dx_Wscale, const float* __restrict__ dx_bscale,
    const float* __restrict__ h1_0,
    const float* __restrict__ h2s,
    float* __restrict__ out)
{
    __shared__ float sWt[WT_TOTAL];     // cached weights/biases (33.6 KB)
    __shared__ float sH1[16 * 12];      // h1, K padded 10->12 with zeros
    __shared__ float sG1[16 * DDZC];    // relu hidden of dz MLP
    __shared__ float sZ[16 * DZC];      // sampled z_t (A for gi1)
    __shared__ float sEps[16 * DZC];    // eps slab for current step
    __shared__ float sH2[16 * DH2C];    // h2_shift slab for current step
    __shared__ float sGi[16 * GPAD];    // gi1 gates
    __shared__ float sGh[16 * GPAD];    // gh1 gates
    __shared__ float sCat[16 * (DH1C + DH2C)];
    __shared__ float sG2[16 * DDXC];    // relu hidden of dx MLP

    const int tid  = (int)threadIdx.x;  // 0..127
    const int lane = tid & 31;
    const int wave = tid >> 5;          // 0..3 (wave-uniform)
    const int b0 = (int)blockIdx.x * 16;

    // ---- one-time weight preload into LDS ----
    ldcpy(sWt + OFF_DZW1,   dz_W1,   3 * DH1C * DH1C * 2, tid, 128); // 600
    ldcpy(sWt + OFF_DZB1,   dz_b1,   DDZC,                tid, 128);
    ldcpy(sWt + OFF_DZWLOC, dz_Wloc, DZC * DDZC,          tid, 128);
    ldcpy(sWt + OFF_DZBLOC, dz_bloc, DZC,                 tid, 128);
    ldcpy(sWt + OFF_DZBSC,  dz_bscale, DZC,               tid, 128);
    ldcpy(sWt + OFF_WIH1,   W_ih1,   G1W * DZC,           tid, 128);
    ldcpy(sWt + OFF_BIH1,   b_ih1,   G1W,                 tid, 128);
    ldcpy(sWt + OFF_WHH1,   W_hh1,   G1W * DH1C,          tid, 128);
    ldcpy(sWt + OFF_BHH1,   b_hh1,   G1W,                 tid, 128);
    ldcpy(sWt + OFF_DXW1,   dx_W1,   DDXC * (DH1C + DH2C), tid, 128);
    ldcpy(sWt + OFF_DXB1,   dx_b1,   DDXC,                tid, 128);
    ldcpy(sWt + OFF_DXWLOC, dx_Wloc, DXC * DDXC,          tid, 128);
    ldcpy(sWt + OFF_DXBLOC, dx_bloc, DXC,                 tid, 128);
    ldcpy(sWt + OFF_DXBSC,  dx_bscale, DXC,               tid, 128);
    for (int idx = tid; idx < 16 * 12; idx += 128) {
        const int j = idx % 12;
        sH1[idx] = (j < DH1C) ? h1_0[j] : 0.0f;
    }
    __syncthreads();

    for (int t = 0; t < TT; ++t) {
        // ---- stage 1: waves 0-1 compute dz hidden GEMM tiles;
        //      wave 2 prefetches h2_shift slab; waves 2-3 prefetch eps slab ----
        if (wave < 2) {
            gemm_tile(sWt + OFF_DZW1, sWt + OFF_DZB1, sH1, 12, sG1, DDZC,
                      DH1C, 12, wave * 16, DDZC, lane);
        } else {
            if (wave == 2) {
                for (int idx = lane; idx < 16 * DH2C; idx += 32) {
                    const int m = idx / DH2C, j = idx % DH2C;
                    sH2[idx] = h2s[((size_t)(b0 + m) * TT + t) * DH2C + j];
                }
            }
            for (int idx = (wave - 2) * 32 + lane; idx < 16 * DZC; idx += 64) {
                const int m = idx / DZC, j = idx % DZC;
                sEps[idx] = eps[((size_t)(b0 + m) * TT + t) * DZC + j];
            }
        }
        __syncthreads();
        for (int idx = tid; idx < 16 * DDZC; idx += 128)
            sG1[idx] = fmaxf(sG1[idx], 0.0f);
        __syncthreads();

        // ---- stage 2: dz heads, 7 N-tiles round-robin over 4 waves ----
        for (int tile = wave; tile * 16 < DZC; tile += 4)
            dz_head_tile(sWt + OFF_DZWLOC, sWt + OFF_DZBLOC, dz_Wscale,
                         sWt + OFF_DZBSC, sG1, sEps, sZ, out, b0, t, tile * 16, lane);
        __syncthreads();

        // ---- stage 3: gi1 (waves 0-1) concurrent with gh1 (waves 2-3) ----
        if (wave < 2)
            gemm_tile(sWt + OFF_WIH1, sWt + OFF_BIH1, sZ, DZC, sGi, GPAD,
                      DZC, DZC, wave * 16, G1W, lane);
        else
            gemm_tile(sWt + OFF_WHH1, sWt + OFF_BHH1, sH1, 12, sGh, GPAD,
                      DH1C, 12, (wave - 2) * 16, G1W, lane);
        __syncthreads();

        // ---- GRU combine (PyTorch gate order r, z, n), h1 update in place ----
        for (int idx = tid; idx < 16 * DH1C; idx += 128) {
            const int m = idx / DH1C, j = idx % DH1C;
            const float rg = sigmoidf_(sGi[m * GPAD + j] + sGh[m * GPAD + j]);
            const float ug = sigmoidf_(sGi[m * GPAD + DH1C + j] + sGh[m * GPAD + DH1C + j]);
            const float ng = tanhf(sGi[m * GPAD + 2 * DH1C + j] + rg * sGh[m * GPAD + 2 * DH1C + j]);
            sH1[m * 12 + j] = (1.0f - ug) * ng + ug * sH1[m * 12 + j];
        }
        __syncthreads();

        // ---- concat [h1n, h2_shift[t]] (both already in LDS) ----
        for (int idx = tid; idx < 16 * (DH1C + DH2C); idx += 128) {
            const int m = idx / (DH1C + DH2C), j = idx % (DH1C + DH2C);
            sCat[idx] = (j < DH1C) ? sH1[m * 12 + j] : sH2[m * DH2C + (j - DH1C)];
        }
        __syncthreads();

        // ---- stage 4: dx hidden GEMM (waves 0-1) ----
        if (wave < 2)
            gemm_tile(sWt + OFF_DXW1, sWt + OFF_DXB1, sCat, DH1C + DH2C, sG2, DDXC,
                      DH1C + DH2C, DH1C + DH2C, wave * 16, DDXC, lane);
        __syncthreads();
        for (int idx = tid; idx < 16 * DDXC; idx += 128)
            sG2[idx] = fmaxf(sG2[idx], 0.0f);
        __syncthreads();

        // ---- stage 5: dx heads, 7 N-tiles round-robin over 4 waves ----
        for (int tile = wave; tile * 16 < DXC; tile += 4)
            dx_head_tile(sWt + OFF_DXWLOC, sWt + OFF_DXBLOC, dx_Wscale,
                         sWt + OFF_DXBSC, sG2, out, b0, t, tile * 16, lane);
        __syncthreads();
    }
}

// -----------------------------------------------------------------------------
extern "C" void kernel_launch(void* const* d_in, const int* in_sizes, int n_in,
                              void* d_out, int out_size, void* d_ws, size_t ws_size,
                              hipStream_t stream) {
    const float* x         = (const float*)d_in[0];
    const float* eps       = (const float*)d_in[1];
    const float* W_ih1     = (const float*)d_in[2];
    const float* W_hh1     = (const float*)d_in[3];
    const float* b_ih1     = (const float*)d_in[4];
    const float* b_hh1     = (const float*)d_in[5];
    const float* W_ih2     = (const float*)d_in[6];
    const float* W_hh2     = (const float*)d_in[7];
    const float* b_ih2     = (const float*)d_in[8];
    const float* b_hh2     = (const float*)d_in[9];
    const float* dz_W1     = (const float*)d_in[10];
    const float* dz_b1     = (const float*)d_in[11];
    const float* dz_Wloc   = (const float*)d_in[12];
    const float* dz_bloc   = (const float*)d_in[13];
    const float* dz_Wscale = (const float*)d_in[14];
    const float* dz_bscale = (const float*)d_in[15];
    const float* dx_W1     = (const float*)d_in[16];
    const float* dx_b1     = (const float*)d_in[17];
    const float* dx_Wloc   = (const float*)d_in[18];
    const float* dx_bloc   = (const float*)d_in[19];
    const float* dx_Wscale = (const float*)d_in[20];
    const float* dx_bscale = (const float*)d_in[21];
    const float* h1_0      = (const float*)d_in[22];
    const float* h2_0      = (const float*)d_in[23];

    float* out = (float*)d_out;
    float* gi2 = (float*)d_ws;                            // B*T*32 floats
    float* h2s = gi2 + (size_t)BB * TT * GPAD;            // B*T*10 floats

    // Phase 1: parallel WMMA GEMM for gi2 = x @ W_ih2^T + b_ih2
    k_gi2<<<1024, 256, 0, stream>>>(x, W_ih2, b_ih2, gi2);

    // Phase 2: batch-parallel GRU2 scan producing shifted h2 sequence
    k_gru2<<<2, 128, 0, stream>>>(gi2, W_hh2, b_hh2, h2_0, h2s);

    // Phase 3: sequential rollout, 4 waves per 16-row batch tile
    k_main<<<BB / 16, 128, 0, stream>>>(eps,
        W_ih1, W_hh1, b_ih1, b_hh1,
        dz_W1, dz_b1, dz_Wloc, dz_bloc, dz_Wscale, dz_bscale,
        dx_W1, dx_b1, dx_Wloc, dx_bloc, dx_Wscale, dx_bscale,
        h1_0, h2s, out);
}